// GNN_28226525070351
// MI455X (gfx1250) — compile-verified
//
#include <hip/hip_runtime.h>
#include <hip/hip_bf16.h>

typedef __attribute__((ext_vector_type(2))) float v2f;
typedef __attribute__((ext_vector_type(8))) float v8f;

#define N_NODES 50000
#define N_EDGES 800000
#define IN_C 10
#define HID_C 64
#define OUT_C 10
#define ROW_TILES 3125  // 50000 / 16

// ---------------- init kernels ----------------

__global__ void gnn_zero4_kernel(float4* __restrict__ p, int n4) {
    int i = blockIdx.x * blockDim.x + threadIdx.x;
    if (i < n4) p[i] = make_float4(0.f, 0.f, 0.f, 0.f);
}

__global__ void gnn_init_out_kernel(float* __restrict__ out, const float* __restrict__ b2) {
    int i = blockIdx.x * blockDim.x + threadIdx.x;
    if (i < N_NODES * OUT_C) out[i] = b2[i % OUT_C];
}

// ---------------- GEMM1: m1 = x @ W1  (bias deferred past aggregation) ----------------
// One block = 4 waves; block b covers rows [16b, 16b+16); wave w covers cols [16w, 16w+16).
// K = 10, padded to 12 -> 3 chained V_WMMA_F32_16X16X4_F32.
__global__ void gnn_gemm1_wmma(const float* __restrict__ x, const float* __restrict__ W1,
                               float* __restrict__ m1) {
    const int wave = threadIdx.x >> 5;
    const int lane = threadIdx.x & 31;
    const int half = lane >> 4;     // 0: K pair {0,1}, 1: K pair {2,3} within a step
    const int l16  = lane & 15;
    const int rt   = blockIdx.x;    // 0..3124
    const int row  = rt * 16 + l16; // A row for this lane
    const int n    = wave * 16 + l16;

    v8f c = {};
#pragma unroll
    for (int ks = 0; ks < 3; ++ks) {
        const int k0 = ks * 4 + half * 2;
        v2f a, b;
        a.x = (k0     < IN_C) ? x[row * IN_C + k0]     : 0.0f;
        a.y = (k0 + 1 < IN_C) ? x[row * IN_C + k0 + 1] : 0.0f;
        b.x = (k0     < IN_C) ? W1[(k0)     * HID_C + n] : 0.0f;
        b.y = (k0 + 1 < IN_C) ? W1[(k0 + 1) * HID_C + n] : 0.0f;
        c = __builtin_amdgcn_wmma_f32_16x16x4_f32(false, a, false, b, (short)0, c,
                                                  false, false);
    }
#pragma unroll
    for (int v = 0; v < 8; ++v) {
        const int m = rt * 16 + v + 8 * half;       // D layout: row = v + 8*(lane/16)
        m1[m * HID_C + wave * 16 + l16] = c[v];
    }
}

// ---------------- Edge scatter layer 1: agg1[dst] += m1[src], 64 ch ----------------
// 16 threads per edge; each handles 4 channels: one b128 load + 4 f32 L2 atomics.
__global__ void gnn_scatter1(const long long* __restrict__ src, const long long* __restrict__ dst,
                             const float* __restrict__ m1, float* __restrict__ agg1) {
    int tid = blockIdx.x * blockDim.x + threadIdx.x;     // over N_EDGES*16 = 12.8M
    if (tid >= N_EDGES * 16) return;
    const int e  = tid >> 4;
    const int c4 = (tid & 15) << 2;                      // 0,4,...,60
    const int s = (int)src[e];
    const int d = (int)dst[e];
    const float4 v = *(const float4*)(m1 + s * HID_C + c4);
    float* p = agg1 + d * HID_C + c4;
    unsafeAtomicAdd(p + 0, v.x);
    unsafeAtomicAdd(p + 1, v.y);
    unsafeAtomicAdd(p + 2, v.z);
    unsafeAtomicAdd(p + 3, v.w);
}

// ---------------- GEMM2: m2 = relu(agg1 + b1) @ W2  (bias+relu fused into A load) -----
// One block = 4 waves; wave handles one 16-row tile; N=10 padded to 16; K=64 -> 16 WMMAs.
// m2 stored padded [N_NODES, 16]; padded cols are exact zero.
__global__ void gnn_gemm2_wmma(const float* __restrict__ agg1, const float* __restrict__ b1,
                               const float* __restrict__ W2, float* __restrict__ m2) {
    const int wave = threadIdx.x >> 5;
    const int lane = threadIdx.x & 31;
    const int half = lane >> 4;
    const int l16  = lane & 15;
    const int rt   = blockIdx.x * 4 + wave;
    if (rt >= ROW_TILES) return;
    const int row = rt * 16 + l16;

    v8f c = {};
#pragma unroll
    for (int ks = 0; ks < 16; ++ks) {
        const int k0 = ks * 4 + half * 2;
        v2f a, b;
        float h0 = agg1[row * HID_C + k0]     + b1[k0];
        float h1 = agg1[row * HID_C + k0 + 1] + b1[k0 + 1];
        a.x = h0 > 0.0f ? h0 : 0.0f;
        a.y = h1 > 0.0f ? h1 : 0.0f;
        b.x = (l16 < OUT_C) ? W2[(k0)     * OUT_C + l16] : 0.0f;
        b.y = (l16 < OUT_C) ? W2[(k0 + 1) * OUT_C + l16] : 0.0f;
        c = __builtin_amdgcn_wmma_f32_16x16x4_f32(false, a, false, b, (short)0, c,
                                                  false, false);
    }
#pragma unroll
    for (int v = 0; v < 8; ++v) {
        const int m = rt * 16 + v + 8 * half;
        m2[m * 16 + l16] = c[v];
    }
}

// ---------------- Edge scatter layer 2: out[dst] += m2[src], 10 ch ----------------
// 16 threads per edge (shift/mask indexing); lanes with c >= 10 exit early.
__global__ void gnn_scatter2(const long long* __restrict__ src, const long long* __restrict__ dst,
                             const float* __restrict__ m2, float* __restrict__ out) {
    int tid = blockIdx.x * blockDim.x + threadIdx.x;     // over N_EDGES*16 = 12.8M
    if (tid >= N_EDGES * 16) return;
    const int e = tid >> 4;
    const int c = tid & 15;
    if (c >= OUT_C) return;
    const int s = (int)src[e];
    const int d = (int)dst[e];
    unsafeAtomicAdd(&out[d * OUT_C + c], m2[s * 16 + c]);
}

extern "C" void kernel_launch(void* const* d_in, const int* in_sizes, int n_in,
                              void* d_out, int out_size, void* d_ws, size_t ws_size,
                              hipStream_t stream) {
    const float*     x    = (const float*)d_in[0];
    const long long* ei   = (const long long*)d_in[1];  // int64 [2, N_EDGES]
    const float*     W1   = (const float*)d_in[2];
    const float*     b1   = (const float*)d_in[3];
    const float*     W2   = (const float*)d_in[4];
    const float*     b2   = (const float*)d_in[5];
    float*           out  = (float*)d_out;

    const long long* src = ei;
    const long long* dst = ei + N_EDGES;

    float* ws   = (float*)d_ws;
    float* m1   = ws;                            // [N_NODES, 64]  = 3,200,000 f32
    float* agg1 = ws + (size_t)N_NODES * HID_C;  // [N_NODES, 64]  = 3,200,000 f32
    float* m2   = ws;                            // alias m1 (dead after scatter1), [N_NODES,16]

    // 1) out = broadcast(b2)
    {
        int n = N_NODES * OUT_C;
        gnn_init_out_kernel<<<(n + 255) / 256, 256, 0, stream>>>(out, b2);
    }
    // 2) agg1 = 0  (float4 vectorized; 3,200,000 % 4 == 0)
    {
        int n4 = (N_NODES * HID_C) / 4;
        gnn_zero4_kernel<<<(n4 + 255) / 256, 256, 0, stream>>>((float4*)agg1, n4);
    }
    // 3) m1 = x @ W1
    gnn_gemm1_wmma<<<ROW_TILES, 128, 0, stream>>>(x, W1, m1);
    // 4) agg1[dst] += m1[src]  (4 channels per thread)
    {
        int n = N_EDGES * 16;                              // 12.8M threads
        gnn_scatter1<<<(n + 255) / 256, 256, 0, stream>>>(src, dst, m1, agg1);
    }
    // 5) m2 = relu(agg1 + b1) @ W2   (padded [N,16])
    gnn_gemm2_wmma<<<(ROW_TILES + 3) / 4, 128, 0, stream>>>(agg1, b1, W2, m2);
    // 6) out[dst] += m2[src]
    {
        int n = N_EDGES * 16;                              // 12.8M threads
        gnn_scatter2<<<(n + 255) / 256, 256, 0, stream>>>(src, dst, m2, out);
    }
}